// GC2_4037269258320
// MI455X (gfx1250) — compile-verified
//
#include <hip/hip_runtime.h>
#include <hip/hip_bf16.h>
#include <math.h>

typedef __attribute__((ext_vector_type(2))) float v2f;
typedef __attribute__((ext_vector_type(8))) float v8f;

#define FDIM 128
#define SS_STRIDE 132   // pad so float4 LDS stores stay 16B-aligned; A-read bank = (4m+k)%64 -> conflict-free

// ---------------------------------------------------------------------------
// Kernel 1: h = in_feat * d[row];  agg = 0   (float4 vectorized, b128 traffic)
// ---------------------------------------------------------------------------
__global__ __launch_bounds__(256) void gcn_prescale_kernel(
    const float4* __restrict__ in4, const float* __restrict__ dvec,
    float4* __restrict__ h4, float4* __restrict__ agg4, int total4) {
  int idx = blockIdx.x * 256 + threadIdx.x;
  if (idx < total4) {
    int row = idx >> 5;                    // 32 float4 per 128-float row
    float dv = dvec[row];
    float4 v = in4[idx];
    v.x *= dv; v.y *= dv; v.z *= dv; v.w *= dv;
    h4[idx] = v;
    agg4[idx] = make_float4(0.f, 0.f, 0.f, 0.f);
  }
}

// ---------------------------------------------------------------------------
// Kernel 2: edge scatter-add. One wave per edge; lane moves float4 => full
// 128-float row per edge. Coalesced 512B gather; native no-return fp32
// atomics into agg (both h and agg are 25.6MB -> resident in the 192MB L2).
// ---------------------------------------------------------------------------
__global__ __launch_bounds__(256) void gcn_scatter_kernel(
    const float* __restrict__ h, const long long* __restrict__ esrc,
    const long long* __restrict__ edst, float* __restrict__ agg, int numE) {
  int edge = blockIdx.x * 8 + (threadIdx.x >> 5);   // 8 waves per block
  int lane = threadIdx.x & 31;
  if (edge >= numE) return;                          // wave-uniform branch
  long long s = esrc[edge];
  long long t = edst[edge];
  const float4 v = *(const float4*)(h + (size_t)s * FDIM + lane * 4);
  float* dst = agg + (size_t)t * FDIM + lane * 4;
  unsafeAtomicAdd(dst + 0, v.x);                     // global_atomic_add_f32
  unsafeAtomicAdd(dst + 1, v.y);
  unsafeAtomicAdd(dst + 2, v.z);
  unsafeAtomicAdd(dst + 3, v.w);
}

// ---------------------------------------------------------------------------
// Kernel 3: fused support + GEMM + residual via V_WMMA_F32_16X16X4_F32.
//   S = (1-a)*agg*d + a*h0      (float4-staged into LDS, 16x128 per block)
//   acc = S@W  (32 chained WMMAs, C0 = 0)
//   out = theta*acc + (1-theta)*S   (residual FMA from LDS; no division)
// 8 waves/block; wave w owns the 16x16 output tile at columns w*16.
// ---------------------------------------------------------------------------
__global__ __launch_bounds__(256) void gcn_gemm_kernel(
    const float* __restrict__ agg, const float* __restrict__ h0,
    const float* __restrict__ dvec, const float* __restrict__ weight,
    const float* __restrict__ lamda, const float* __restrict__ alpha,
    const int* __restrict__ lval, float* __restrict__ out, int numRows) {
  __shared__ __align__(16) float sS[16 * SS_STRIDE];

  const int tid  = threadIdx.x;
  const int lane = tid & 31;
  const int wave = tid >> 5;        // 0..7
  const int row0 = blockIdx.x * 16;
  const int n0   = wave * 16;
  const int half = lane >> 4;       // 0 | 1
  const int ln   = lane & 15;       // 0..15

  const float a     = alpha[0];
  const float theta = __logf(lamda[0] / (float)lval[0] + 1.0f);

  // Stage the 16x128 support tile: 512 float4 elements, 2 per thread (b128).
  for (int i = tid; i < 16 * (FDIM / 4); i += 256) {
    int r  = i >> 5;                 // row within tile
    int c4 = (i & 31) << 2;          // starting column (multiple of 4)
    int g  = row0 + r;
    if (g >= numRows) g = numRows - 1;                 // clamp (keeps EXEC full)
    float dv = dvec[g];
    const float4 av = *(const float4*)(agg + (size_t)g * FDIM + c4);
    const float4 hv = *(const float4*)(h0  + (size_t)g * FDIM + c4);
    float4 sv;
    sv.x = fmaf((1.0f - a), av.x * dv, a * hv.x);
    sv.y = fmaf((1.0f - a), av.y * dv, a * hv.y);
    sv.z = fmaf((1.0f - a), av.z * dv, a * hv.z);
    sv.w = fmaf((1.0f - a), av.w * dv, a * hv.w);
    *(float4*)(&sS[r * SS_STRIDE + c4]) = sv;
  }
  __syncthreads();

  v8f c = {};                        // C0 = 0; residual applied after the chain

  // K loop: 32 x V_WMMA_F32_16X16X4_F32
#pragma unroll
  for (int k0 = 0; k0 < FDIM; k0 += 4) {
    int kk = k0 + 2 * half;
    // A 16x4 fragment: m = lane%16; V0=A[m][kk], V1=A[m][kk+1]  (ds_load_b64)
    v2f af = *(const v2f*)(&sS[ln * SS_STRIDE + kk]);
    // B 4x16 fragment: n = n0+lane%16; V0=W[kk][n], V1=W[kk+1][n]
    // 16 consecutive floats per half-wave; weight (64KB) stays hot in WGP$/L2.
    v2f bf;
    bf.x = weight[(size_t)kk * FDIM + n0 + ln];
    bf.y = weight[(size_t)(kk + 1) * FDIM + n0 + ln];
    c = __builtin_amdgcn_wmma_f32_16x16x4_f32(
        /*neg_a=*/false, af, /*neg_b=*/false, bf,
        /*c_mod=*/(short)0, c, /*reuse_a=*/false, /*reuse_b=*/false);
  }

  // out = theta*acc + (1-theta)*S  (S re-read from LDS; C/D layout m=r+8*half)
  const float omt = 1.0f - theta;
#pragma unroll
  for (int r = 0; r < 8; ++r) {
    int m = r + 8 * half;
    int grow = row0 + m;
    if (grow < numRows) {
      float sv = sS[m * SS_STRIDE + n0 + ln];
      out[(size_t)grow * FDIM + n0 + ln] = fmaf(theta, c[r], omt * sv);
    }
  }
}

// ---------------------------------------------------------------------------
// Launch: in_feat, h0, d, weight, lamda, alpha, edge_src(i64), edge_dst(i64), l
// ---------------------------------------------------------------------------
extern "C" void kernel_launch(void* const* d_in, const int* in_sizes, int n_in,
                              void* d_out, int out_size, void* d_ws, size_t ws_size,
                              hipStream_t stream) {
  const float*     in_feat = (const float*)d_in[0];
  const float*     h0      = (const float*)d_in[1];
  const float*     dvec    = (const float*)d_in[2];
  const float*     weight  = (const float*)d_in[3];
  const float*     lamda   = (const float*)d_in[4];
  const float*     alpha   = (const float*)d_in[5];
  const long long* esrc    = (const long long*)d_in[6];
  const long long* edst    = (const long long*)d_in[7];
  const int*       lval    = (const int*)d_in[8];
  float*           out     = (float*)d_out;

  const int numRows = in_sizes[2];          // N  (d has N elements)
  const int total   = in_sizes[0];          // N*F
  const int numE    = in_sizes[6];          // E

  float* h   = (float*)d_ws;                // N*F floats
  float* agg = h + (size_t)total;           // N*F floats

  const int total4 = total >> 2;
  gcn_prescale_kernel<<<(total4 + 255) / 256, 256, 0, stream>>>(
      (const float4*)in_feat, dvec, (float4*)h, (float4*)agg, total4);
  gcn_scatter_kernel<<<(numE + 7) / 8, 256, 0, stream>>>(
      h, esrc, edst, agg, numE);
  gcn_gemm_kernel<<<(numRows + 15) / 16, 256, 0, stream>>>(
      agg, h0, dvec, weight, lamda, alpha, lval, out, numRows);
}